// TorchFlowLenia_79508434584075
// MI455X (gfx1250) — compile-verified
//
#include <hip/hip_runtime.h>
#include <hip/hip_bf16.h>
#include <math.h>

typedef __attribute__((ext_vector_type(2))) float v2f;
typedef __attribute__((ext_vector_type(8))) float v8f;

#define SX 512
#define SY 512
#define CC 3
#define NKR 10
#define MID 256
#define RW 48            // tap window di,dj in [-48,47] -> 96 per axis
#define WIN 96
#define AW 112           // staged A row width (covers m+dj+kk range [-48,62])

// ---- gfx1250 async global->LDS copy support (guarded; falls back to ld/st) --
#if defined(__has_builtin)
#if __has_builtin(__builtin_amdgcn_global_load_async_to_lds_b32) && \
    __has_builtin(__builtin_amdgcn_global_load_async_to_lds_b64)
#define HAVE_ASYNC_LDS 1
#endif
#endif
#ifndef HAVE_ASYNC_LDS
#define HAVE_ASYNC_LDS 0
#endif

#define GLOBAL_AS __attribute__((address_space(1)))
#define LDS_AS    __attribute__((address_space(3)))
typedef int v2i_vs __attribute__((vector_size(8)));   // matches builtin pointee

__device__ __forceinline__ void wait_async_all() {
#if HAVE_ASYNC_LDS
#if __has_builtin(__builtin_amdgcn_s_wait_asynccnt)
  __builtin_amdgcn_s_wait_asynccnt(0);
#else
  asm volatile("s_wait_asynccnt 0x0" ::: "memory");
#endif
#endif
}

// ---------------------------------------------------------------------------
// Kernel A: truncated-window circular Lenia convolution as implicit GEMM via
// V_WMMA_F32_16X16X4_F32, double-buffered async staging into LDS.
// U[x,y,k] = sum_{di,dj} nK[mid-di, mid-dj, k] * A[(x+di)%SX, (y+dj)%SY, c0[k]]
// ---------------------------------------------------------------------------
__global__ __launch_bounds__(256)
void lenia_conv_wmma(const float* __restrict__ A, const float* __restrict__ nK,
                     const int* __restrict__ c0, float* __restrict__ U)
{
  __shared__ __align__(16) float Ash[2][3 * 8 * AW];   // [buf][ch][xrow][t]
  __shared__ __align__(16) float Braw[2][WIN * NKR];   // [buf][t=ky-209][n]

  const int tid  = threadIdx.x;
  const int y0   = blockIdx.x * 16;     // 32 y-tiles
  const int xb   = blockIdx.y * 8;      // 64 x-groups of 8 rows
  const int wv   = tid >> 5;            // wave id -> x row
  const int l    = tid & 31;
  const int half = l >> 4;              // lane half selects K sub-rows {0,1}/{2,3}
  const int mrow = l & 15;              // M row for A-frag, N column for B-frag

  // per-lane channel one-hot for the B-fragment column this lane owns
  const int cn      = (mrow < NKR) ? c0[mrow] : -1;
  const int nclamp  = (mrow < NKR) ? mrow : (NKR - 1);
  float mask[3];
  mask[0] = (cn == 0) ? 1.f : 0.f;
  mask[1] = (cn == 1) ? 1.f : 0.f;
  mask[2] = (cn == 2) ? 1.f : 0.f;

  v8f acc = {};                          // 16x16 f32 C/D fragment

  // stage tap-row `di` into buffer `buf` (async when available)
  auto stage = [&](int di, int buf) {
    // shifted A rows for all 3 channels (scattered b32 copies: handles wrap)
    for (int i = tid; i < 3 * 8 * AW; i += 256) {
      int c   = i / (8 * AW);
      int rem = i - c * (8 * AW);
      int w   = rem / AW;
      int t   = rem - w * AW;
      int xr  = (xb + w + di) & (SX - 1);
      int yr  = (y0 + t - RW) & (SY - 1);
      const float* gp = A + ((size_t)(xr * SY + yr)) * CC + c;
#if HAVE_ASYNC_LDS
      __builtin_amdgcn_global_load_async_to_lds_b32(
          (GLOBAL_AS int*)gp, (LDS_AS int*)&Ash[buf][i], 0, 0);
#else
      Ash[buf][i] = *gp;
#endif
    }
    // kernel row (mid-di, ky=209..304, n=0..9): one contiguous 960-float span
    const float* src = nK + ((size_t)((MID - di) * SY + (MID - RW + 1))) * NKR;
    for (int i = tid; i < (WIN * NKR) / 2; i += 256) {     // 480 b64 copies
#if HAVE_ASYNC_LDS
      __builtin_amdgcn_global_load_async_to_lds_b64(
          (GLOBAL_AS v2i_vs*)(src + 2 * i),
          (LDS_AS v2i_vs*)&Braw[buf][2 * i], 0, 0);
#else
      Braw[buf][2 * i]     = src[2 * i];
      Braw[buf][2 * i + 1] = src[2 * i + 1];
#endif
    }
  };

  stage(-RW, 0);
  wait_async_all();
  __syncthreads();

  int buf = 0;
  for (int di = -RW; di < RW; ++di) {
    if (di + 1 < RW) stage(di + 1, buf ^ 1);   // prefetch next tap row

    const float* __restrict__ Ab = &Ash[buf][0];
    const float* __restrict__ Bb = &Braw[buf][0];

    #pragma unroll 4
    for (int cnk = 0; cnk < WIN / 4; ++cnk) {
      const int tb = cnk * 4 + 2 * half;           // this lane's K sub-rows
      // raw kernel weights for K rows tb, tb+1 (djIdx -> t = 95 - djIdx)
      const float braw0 = Bb[(95 - tb) * NKR + nclamp];
      const float braw1 = Bb[(94 - tb) * NKR + nclamp];
      #pragma unroll
      for (int ch = 0; ch < 3; ++ch) {
        v2f a, b;
        const float* ap = Ab + ch * (8 * AW) + wv * AW + mrow + tb;
        a.x = ap[0];                 // A[m=mrow][K=tb]
        a.y = ap[1];                 // A[m=mrow][K=tb+1]
        b.x = braw0 * mask[ch];      // zero unless c0[n]==ch
        b.y = braw1 * mask[ch];
        acc = __builtin_amdgcn_wmma_f32_16x16x4_f32(
                  false, a, false, b, (short)0, acc, false, false);
      }
    }

    wait_async_all();
    __syncthreads();
    buf ^= 1;
  }

  const int x = xb + wv;
  if (mrow < NKR) {
    #pragma unroll
    for (int v = 0; v < 8; ++v) {
      int m = v + 8 * half;              // C layout: VGPR v, lane-half -> M
      U[((size_t)(x * SY + (y0 + m))) * NKR + mrow] = acc[v];
    }
  }
}

// ---------------------------------------------------------------------------
// Kernel B: growth function, one-hot gather into target channels, channel sum
// ---------------------------------------------------------------------------
__global__ __launch_bounds__(256)
void growth_gather(const float* __restrict__ U, const float* __restrict__ A,
                   const float* __restrict__ gm, const float* __restrict__ gs,
                   const float* __restrict__ gh, const int* __restrict__ c1,
                   float* __restrict__ Uc, float* __restrict__ Asum)
{
  int p = blockIdx.x * 256 + threadIdx.x;
  if (p >= SX * SY) return;
  float uc0 = 0.f, uc1 = 0.f, uc2 = 0.f;
  #pragma unroll
  for (int k = 0; k < NKR; ++k) {
    float u  = U[p * NKR + k];
    float d  = u - gm[k];
    float sk = gs[k];
    float g  = (2.0f * __expf(-d * d / (2.0f * sk * sk)) - 1.0f) * gh[k];
    uc0 += g * (float)c1[k * 3 + 0];
    uc1 += g * (float)c1[k * 3 + 1];
    uc2 += g * (float)c1[k * 3 + 2];
  }
  Uc[p * 3 + 0] = uc0; Uc[p * 3 + 1] = uc1; Uc[p * 3 + 2] = uc2;
  Asum[p] = A[p * 3 + 0] + A[p * 3 + 1] + A[p * 3 + 2];
}

// ---------------------------------------------------------------------------
// Kernel C: Sobel flow field, alpha blend, clipped advection target mu
// ---------------------------------------------------------------------------
__global__ __launch_bounds__(256)
void flow_mu(const float* __restrict__ Uc, const float* __restrict__ Asum,
             const float* __restrict__ A, float* __restrict__ mu)
{
  int p = blockIdx.x * 256 + threadIdx.x;
  if (p >= SX * SY) return;
  int x = p >> 9, y = p & 511;
  int xm = (x - 1) & 511, xp = (x + 1) & 511;
  int ym = (y - 1) & 511, yp = (y + 1) & 511;
  int i00 = xm * SY + ym, i01 = xm * SY + y, i02 = xm * SY + yp;
  int i10 = x  * SY + ym,                     i12 = x * SY + yp;
  int i20 = xp * SY + ym, i21 = xp * SY + y, i22 = xp * SY + yp;

  float a00 = Asum[i00], a01 = Asum[i01], a02 = Asum[i02];
  float a10 = Asum[i10], a12 = Asum[i12];
  float a20 = Asum[i20], a21 = Asum[i21], a22 = Asum[i22];
  float gAx = ((a20 + 2.f * a21 + a22) - (a00 + 2.f * a01 + a02)) * 0.125f;
  float gAy = ((a02 + 2.f * a12 + a22) - (a00 + 2.f * a10 + a20)) * 0.125f;

  const float ma = 5.0f - 0.65f;     // DD - SIGMA
  #pragma unroll
  for (int c = 0; c < 3; ++c) {
    float n00 = Uc[i00 * 3 + c], n01 = Uc[i01 * 3 + c], n02 = Uc[i02 * 3 + c];
    float n10 = Uc[i10 * 3 + c],                          n12 = Uc[i12 * 3 + c];
    float n20 = Uc[i20 * 3 + c], n21 = Uc[i21 * 3 + c], n22 = Uc[i22 * 3 + c];
    float gx = ((n20 + 2.f * n21 + n22) - (n00 + 2.f * n01 + n02)) * 0.125f;
    float gy = ((n02 + 2.f * n12 + n22) - (n00 + 2.f * n10 + n20)) * 0.125f;

    float av    = A[p * 3 + c];
    float alpha = fminf(av * av, 1.0f);          // clip((A/1)^2, 0, 1)
    float Fx = gx * (1.f - alpha) - gAx * alpha;
    float Fy = gy * (1.f - alpha) - gAy * alpha;
    Fx = fminf(fmaxf(Fx, -ma), ma);
    Fy = fminf(fmaxf(Fy, -ma), ma);
    mu[(p * 2 + 0) * 3 + c] = (float)x + 0.5f + 0.2f * Fx;
    mu[(p * 2 + 1) * 3 + c] = (float)y + 0.5f + 0.2f * Fy;
  }
}

// ---------------------------------------------------------------------------
// Kernel D: mass-conserving reintegration over 121 shifts, LDS halo tile
// ---------------------------------------------------------------------------
#define HALO 5
#define TD 16
#define TW (TD + 2 * HALO)   // 26

__global__ __launch_bounds__(256)
void reintegrate(const float* __restrict__ A, const float* __restrict__ mu,
                 float* __restrict__ out)
{
  __shared__ float sA[TW * TW * 3];
  __shared__ float sMu[TW * TW * 6];
  const int tid = threadIdx.x;
  const int gx0 = blockIdx.x * TD, gy0 = blockIdx.y * TD;

  for (int i = tid; i < TW * TW; i += 256) {
    int lx = i / TW, ly = i - (i / TW) * TW;
    int x = (gx0 + lx - HALO) & 511;
    int y = (gy0 + ly - HALO) & 511;
    int p = x * SY + y;
    sA[i * 3 + 0] = A[p * 3 + 0];
    sA[i * 3 + 1] = A[p * 3 + 1];
    sA[i * 3 + 2] = A[p * 3 + 2];
    #pragma unroll
    for (int j = 0; j < 6; ++j) sMu[i * 6 + j] = mu[p * 6 + j];
  }
  __syncthreads();

  const int tx = tid >> 4, ty = tid & 15;
  const int x = gx0 + tx, y = gy0 + ty;
  const float px = (float)x + 0.5f, py = (float)y + 0.5f;
  const float inv_norm = 1.0f / (4.0f * 0.65f * 0.65f);
  float acc0 = 0.f, acc1 = 0.f, acc2 = 0.f;

  for (int dx = -HALO; dx <= HALO; ++dx)
    for (int dy = -HALO; dy <= HALO; ++dy) {
      int li = (tx - dx + HALO) * TW + (ty - dy + HALO);
      const float* mp = &sMu[li * 6];
      const float* ap = &sA[li * 3];
      #pragma unroll
      for (int c = 0; c < 3; ++c) {
        float szx = 0.5f - fabsf(px - mp[c])     + 0.65f;
        float szy = 0.5f - fabsf(py - mp[3 + c]) + 0.65f;
        float cx = fminf(fmaxf(szx, 0.0f), 1.0f);   // clip_hi = min(1, 2*sigma)=1
        float cy = fminf(fmaxf(szy, 0.0f), 1.0f);
        float area = cx * cy * inv_norm;
        float v = ap[c] * area;
        if (c == 0) acc0 += v; else if (c == 1) acc1 += v; else acc2 += v;
      }
    }
  int p = x * SY + y;
  out[p * 3 + 0] = acc0; out[p * 3 + 1] = acc1; out[p * 3 + 2] = acc2;
}

// ---------------------------------------------------------------------------
extern "C" void kernel_launch(void* const* d_in, const int* in_sizes, int n_in,
                              void* d_out, int out_size, void* d_ws, size_t ws_size,
                              hipStream_t stream) {
  (void)in_sizes; (void)n_in; (void)out_size; (void)ws_size;
  const float* A   = (const float*)d_in[0];   // (512,512,3)
  const float* nK  = (const float*)d_in[1];   // (512,512,10)
  const float* gm  = (const float*)d_in[2];   // (10)
  const float* gs  = (const float*)d_in[3];   // (10)
  const float* gh  = (const float*)d_in[4];   // (10)
  const int*   c0  = (const int*)d_in[5];     // (10)
  const int*   c1  = (const int*)d_in[6];     // (10,3)

  float* ws   = (float*)d_ws;
  float* U    = ws;                            // 512*512*10
  float* Uc   = U  + SX * SY * NKR;            // 512*512*3
  float* Asum = Uc + SX * SY * 3;              // 512*512
  float* mu   = Asum + SX * SY;                // 512*512*6

  lenia_conv_wmma<<<dim3(SY / 16, SX / 8), 256, 0, stream>>>(A, nK, c0, U);
  growth_gather<<<(SX * SY) / 256, 256, 0, stream>>>(U, A, gm, gs, gh, c1, Uc, Asum);
  flow_mu<<<(SX * SY) / 256, 256, 0, stream>>>(Uc, Asum, A, mu);
  reintegrate<<<dim3(SX / TD, SY / TD), 256, 0, stream>>>(A, mu, (float*)d_out);
}